// ChebBen1BN_71159018160656
// MI455X (gfx1250) — compile-verified
//
#include <hip/hip_runtime.h>
#include <hip/hip_bf16.h>

// ---------------------------------------------------------------------------
// ChebConv(K=3, sym norm, lambda_max=2) + BatchNorm1d for MI455X (gfx1250).
// f32 end-to-end; matmuls via V_WMMA_F32_16X16X4_F32 (wave32 WMMA).
// ---------------------------------------------------------------------------

#define NNODES 50000
#define NEDGES 800000
#define DFEAT  96
#define ND     (NNODES * DFEAT)          // 4,800,000
#define NROWB  (NNODES / 16)             // 3125 row tiles (exact)
#define BN_EPS 1e-5f

typedef __attribute__((ext_vector_type(2))) float v2f;
typedef __attribute__((ext_vector_type(8))) float v8f;

// ---------------------------------------------------------------------------
// K1: init workspace: T1 = 0, T2 = -x, deg = 0, stats = 0
// ---------------------------------------------------------------------------
__global__ void cheb_init_kernel(const float* __restrict__ x,
                                 float* __restrict__ T1,
                                 float* __restrict__ T2,
                                 float* __restrict__ deg,
                                 float* __restrict__ stats) {
    int i = blockIdx.x * blockDim.x + threadIdx.x;
    if (i < ND) {
        T1[i] = 0.0f;
        T2[i] = -x[i];
    }
    if (i < NNODES) deg[i] = 0.0f;
    if (i < 192)    stats[i] = 0.0f;
}

// ---------------------------------------------------------------------------
// K2: out-degree over non-self-loop edges (row side)
// ---------------------------------------------------------------------------
__global__ void cheb_degree_kernel(const int* __restrict__ ei,
                                   float* __restrict__ deg) {
    int e = blockIdx.x * blockDim.x + threadIdx.x;
    if (e >= NEDGES) return;
    int r = ei[e];
    int c = ei[NEDGES + e];
    if (r != c) atomicAdd(&deg[r], 1.0f);
}

// ---------------------------------------------------------------------------
// K3: deg -> deg^{-1/2} in place (0 where deg == 0)
// ---------------------------------------------------------------------------
__global__ void cheb_dis_kernel(float* __restrict__ deg) {
    int n = blockIdx.x * blockDim.x + threadIdx.x;
    if (n >= NNODES) return;
    float d = deg[n];
    deg[n] = (d > 0.0f) ? rsqrtf(d) : 0.0f;
}

// ---------------------------------------------------------------------------
// K4: per-edge norm = -dis[row]*dis[col], 0 on self loops
// ---------------------------------------------------------------------------
__global__ void cheb_norm_kernel(const int* __restrict__ ei,
                                 const float* __restrict__ dis,
                                 float* __restrict__ norm) {
    int e = blockIdx.x * blockDim.x + threadIdx.x;
    if (e >= NEDGES) return;
    int r = ei[e];
    int c = ei[NEDGES + e];
    norm[e] = (r != c) ? (-dis[r] * dis[c]) : 0.0f;
}

// ---------------------------------------------------------------------------
// K5/K6: scatter: dst[col,:] += scale*norm[e] * src[row,:]
// One wave per edge, each lane owns 3 consecutive features (96 = 32*3).
// Gathers are coalesced 384B bursts; feature matrices are L2-resident.
// ---------------------------------------------------------------------------
__global__ void cheb_scatter_kernel(const float* __restrict__ src,
                                    const int* __restrict__ ei,
                                    const float* __restrict__ norm,
                                    float* __restrict__ dst,
                                    float scale) {
    int t = blockIdx.x * blockDim.x + threadIdx.x;
    int e    = t >> 5;
    int lane = t & 31;
    if (e >= NEDGES) return;
    float w = norm[e] * scale;
    if (w == 0.0f) return;                    // self loop / isolated node
    int r = ei[e];
    int c = ei[NEDGES + e];
    int d0 = lane * 3;
    const float* s = src + r * DFEAT + d0;
    float*       d = dst + c * DFEAT + d0;
    atomicAdd(&d[0], w * s[0]);
    atomicAdd(&d[1], w * s[1]);
    atomicAdd(&d[2], w * s[2]);
}

// ---------------------------------------------------------------------------
// K7: PRE = x@W0 + T1@W1 + T2@W2 + bias via V_WMMA_F32_16X16X4_F32.
// Block = one 16-row tile, 6 waves (192 thr), wave w -> cols [16w,16w+16).
// A tiles (3 x 16x96) staged in LDS with padded stride; 72 WMMAs / wave.
// ISA lane layouts (CDNA5 ISA):
//   A 16x4 f32 : lane L -> M=L%16, VGPR j -> K = j + 2*(L/16)
//   B 4x16 f32 : lane L -> N=L%16, VGPR j -> K = j + 2*(L/16)
//   C/D 16x16  : lane L -> N=L%16, VGPR v -> M = v + 8*(L/16)
// ---------------------------------------------------------------------------
#define APAD 100   // LDS row stride in floats (bank-conflict padding)

__global__ void __launch_bounds__(192)
cheb_gemm_wmma_kernel(const float* __restrict__ x,
                      const float* __restrict__ T1,
                      const float* __restrict__ T2,
                      const float* __restrict__ Wt,     // [3,96,96]
                      const float* __restrict__ bias,   // [96]
                      float* __restrict__ pre) {        // [N,96]
    __shared__ float As[3][16][APAD];

    const int rowBase = blockIdx.x * 16;

    // Cooperative stage of the three 16x96 A tiles into LDS.
    for (int i = threadIdx.x; i < 3 * 16 * DFEAT; i += 192) {
        int mat = i / (16 * DFEAT);
        int rem = i - mat * (16 * DFEAT);
        int r   = rem / DFEAT;
        int c   = rem - r * DFEAT;
        const float* s = (mat == 0) ? x : ((mat == 1) ? T1 : T2);
        As[mat][r][c] = s[(rowBase + r) * DFEAT + c];
    }
    __syncthreads();

    const int wave    = threadIdx.x >> 5;   // 0..5
    const int lane    = threadIdx.x & 31;
    const int m       = lane & 15;          // M (for A) / N (for B,C,D)
    const int h       = lane >> 4;          // K-half selector
    const int colBase = wave * 16;

    v8f acc = {};
    for (int mat = 0; mat < 3; ++mat) {
        const float* Wm = Wt + mat * DFEAT * DFEAT;
        for (int kk = 0; kk < DFEAT; kk += 4) {
            const int k0 = kk + 2 * h;
            v2f a;
            a.x = As[mat][m][k0];
            a.y = As[mat][m][k0 + 1];
            v2f b;
            b.x = Wm[(k0    ) * DFEAT + colBase + m];
            b.y = Wm[(k0 + 1) * DFEAT + colBase + m];
            acc = __builtin_amdgcn_wmma_f32_16x16x4_f32(
                /*neg_a=*/false, a, /*neg_b=*/false, b,
                /*c_mod=*/(short)0, acc, /*reuse_a=*/false, /*reuse_b=*/false);
        }
    }

    const int   col = colBase + m;
    const float bc  = bias[col];
#pragma unroll
    for (int v = 0; v < 8; ++v) {
        int row = rowBase + v + 8 * h;
        pre[row * DFEAT + col] = acc[v] + bc;
    }
}

// ---------------------------------------------------------------------------
// K8: per-feature sum / sumsq (coalesced reads, LDS ds_add_f32 accumulation,
// one 192-wide global atomic flush per block).
// ---------------------------------------------------------------------------
__global__ void cheb_bnstats_kernel(const float* __restrict__ pre,
                                    float* __restrict__ stats) {  // [192]
    __shared__ float ss[192];
    for (int i = threadIdx.x; i < 192; i += blockDim.x) ss[i] = 0.0f;
    __syncthreads();

    const int stride = gridDim.x * blockDim.x;
    for (int i = blockIdx.x * blockDim.x + threadIdx.x; i < ND; i += stride) {
        float v = pre[i];
        int   f = i % DFEAT;
        atomicAdd(&ss[f], v);
        atomicAdd(&ss[DFEAT + f], v * v);
    }
    __syncthreads();
    for (int i = threadIdx.x; i < 192; i += blockDim.x)
        atomicAdd(&stats[i], ss[i]);
}

// ---------------------------------------------------------------------------
// K9: normalize: out = (pre - mean) * rsqrt(var + eps) * gamma + beta
// ---------------------------------------------------------------------------
__global__ void cheb_bnapply_kernel(const float* __restrict__ pre,
                                    const float* __restrict__ stats,
                                    const float* __restrict__ gamma,
                                    const float* __restrict__ beta,
                                    float* __restrict__ out) {
    int i = blockIdx.x * blockDim.x + threadIdx.x;
    if (i >= ND) return;
    int   f     = i % DFEAT;
    const float invN = 1.0f / (float)NNODES;
    float mean  = stats[f] * invN;
    float var   = stats[DFEAT + f] * invN - mean * mean;
    float scale = rsqrtf(var + BN_EPS) * gamma[f];
    out[i] = (pre[i] - mean) * scale + beta[f];
}

// ---------------------------------------------------------------------------
// Host-side launch sequence (all on `stream`, graph-capture safe).
// ---------------------------------------------------------------------------
extern "C" void kernel_launch(void* const* d_in, const int* in_sizes, int n_in,
                              void* d_out, int out_size, void* d_ws, size_t ws_size,
                              hipStream_t stream) {
    const float* x     = (const float*)d_in[0];
    const int*   ei    = (const int*)  d_in[1];
    const float* Wt    = (const float*)d_in[2];
    const float* bias  = (const float*)d_in[3];
    const float* gamma = (const float*)d_in[4];
    const float* beta  = (const float*)d_in[5];
    float*       out   = (float*)d_out;

    // Workspace layout (floats): T1 | T2 | PRE | DIS(=deg) | NORM | STATS
    float* ws    = (float*)d_ws;
    float* T1    = ws;
    float* T2    = T1 + ND;
    float* PRE   = T2 + ND;
    float* DIS   = PRE + ND;
    float* NORM  = DIS + NNODES;
    float* STATS = NORM + NEDGES;

    const int TPB = 256;

    // K1: init
    cheb_init_kernel<<<(ND + TPB - 1) / TPB, TPB, 0, stream>>>(x, T1, T2, DIS, STATS);
    // K2: degree
    cheb_degree_kernel<<<(NEDGES + TPB - 1) / TPB, TPB, 0, stream>>>(ei, DIS);
    // K3: deg -> deg^{-1/2}
    cheb_dis_kernel<<<(NNODES + TPB - 1) / TPB, TPB, 0, stream>>>(DIS);
    // K4: edge norms
    cheb_norm_kernel<<<(NEDGES + TPB - 1) / TPB, TPB, 0, stream>>>(ei, DIS, NORM);
    // K5: T1 = L_hat @ x           (T1 pre-zeroed)
    cheb_scatter_kernel<<<(NEDGES * 32) / TPB, TPB, 0, stream>>>(x,  ei, NORM, T1, 1.0f);
    // K6: T2 = 2 * L_hat @ T1 - x  (T2 pre-initialized to -x)
    cheb_scatter_kernel<<<(NEDGES * 32) / TPB, TPB, 0, stream>>>(T1, ei, NORM, T2, 2.0f);
    // K7: PRE = x@W0 + T1@W1 + T2@W2 + bias  (WMMA f32)
    cheb_gemm_wmma_kernel<<<NROWB, 192, 0, stream>>>(x, T1, T2, Wt, bias, PRE);
    // K8: batch stats
    cheb_bnstats_kernel<<<256, TPB, 0, stream>>>(PRE, STATS);
    // K9: batchnorm apply
    cheb_bnapply_kernel<<<(ND + TPB - 1) / TPB, TPB, 0, stream>>>(PRE, STATS, gamma, beta, out);
}